// SpatialEmbedding_10084583211123
// MI455X (gfx1250) — compile-verified
//
#include <hip/hip_runtime.h>
#include <math.h>

typedef __attribute__((ext_vector_type(2))) float v2f;
typedef __attribute__((ext_vector_type(8))) float v8f;

#define NW      256    // N_WAVES
#define JCHUNK  128    // sources staged in LDS per iteration (8 waves x 16)

// Block: 256 threads (8 wave32), handles one batch b and 16 target rows.
// Wave w produces the (16 targets x 16 sources) r/amp subtile for its slice of
// the j-chunk using V_WMMA_F32_16X16X4_F32; every thread then consumes the
// whole 128-j chunk for its own wavelength m = 32*w + lane.
__global__ __launch_bounds__(256) void spatial_embed_kernel(
    const float* __restrict__ coords, const unsigned char* __restrict__ pad,
    float* __restrict__ out, int N) {
  __shared__ __align__(16) float2 tile[JCHUNK * 16];  // [j_local][i_local] = (r, amp)

  const int b    = blockIdx.y;
  const int i0   = blockIdx.x * 16;
  const int tid  = threadIdx.x;
  const int w    = tid >> 5;
  const int lane = tid & 31;
  const int half = (lane >> 4) & 1;   // lane half selects K (A/B) or M+8 (C/D)
  const int l16  = lane & 15;

  const float*         cb = coords + (size_t)b * N * 3;
  const unsigned char* pb = pad    + (size_t)b * N;

  // wavelength schedule: lam = geomspace(2, 100, 256); k/(2pi) = 1/lam.
  // v_cos_f32/v_sin_f32 compute cos/sin(2*pi*x), so phase input = r / lam.
  const int    m    = w * 32 + lane;
  const double lam  = 2.0 * exp(((double)m / 255.0) * 3.9120230054281460586 /*ln 50*/);
  const float  krev = (float)(1.0 / lam);

  // WMMA A operand (16x4 f32, 2 VGPRs): lanes0-15 hold M=l,(K0,K1); lanes16-31 M=l-16,(K2,K3)
  const int   ia = i0 + l16;
  const float ax = cb[ia * 3 + 0], ay = cb[ia * 3 + 1], az = cb[ia * 3 + 2];
  v2f A;
  A.x = half ? az : ax;     // K=0 / K=2
  A.y = half ? 0.0f : ay;   // K=1 / K=3 (pad component = 0)

  // ||x_i||^2 for the 8 target rows this lane sees in the C/D layout (M = v + 8*half)
  float ni[8];
#pragma unroll
  for (int v = 0; v < 8; ++v) {
    const int   ir = i0 + v + 8 * half;
    const float x = cb[ir * 3 + 0], y = cb[ir * 3 + 1], z = cb[ir * 3 + 2];
    ni[v] = x * x + y * y + z * z;
  }

  float accre[16], accim[16];
#pragma unroll
  for (int q = 0; q < 16; ++q) { accre[q] = 0.0f; accim[q] = 0.0f; }

  for (int jc = 0; jc < N; jc += JCHUNK) {
    // ---------------- producer: wave w fills sources [jc + 16w, jc + 16w + 16) ----------
    const int   j  = jc + w * 16 + l16;
    const float cx = cb[j * 3 + 0], cy = cb[j * 3 + 1], cz = cb[j * 3 + 2];
    v2f Bv;                              // B = -2 * x_j  (4x16, same lane-half K split)
    Bv.x = half ? (-2.0f * cz) : (-2.0f * cx);
    Bv.y = half ? 0.0f : (-2.0f * cy);
    v8f Cz = {};
    // D[i][j] = -2 * (x_i . x_j)
    v8f D = __builtin_amdgcn_wmma_f32_16x16x4_f32(false, A, false, Bv,
                                                  (short)0, Cz, false, false);
    const float nj   = cx * cx + cy * cy + cz * cz;
    const bool  jpad = (pb[j] != 0);
#pragma unroll
    for (int v = 0; v < 8; ++v) {
      const int  ir = i0 + v + 8 * half;
      float      r2 = D[v] + ni[v] + nj;
      const bool pv = (j != ir) && !jpad;          // exclude self + padded sources
      r2 = pv ? fmaxf(r2, 1e-12f) : 1.0f;          // matches reference sq_safe
      const float rs  = __builtin_amdgcn_rsqf(r2);
      const float r   = r2 * rs;                   // sqrt(r2)
      const float amp = pv ? (0.07957747154594767f * rs) : 0.0f;  // 1/(4*pi*r)
      tile[(w * 16 + l16) * 16 + v + 8 * half] = make_float2(r, amp);
    }
    __syncthreads();

    // ---------------- consumer: this thread = wavelength m, all 16 targets --------------
    for (int jl = 0; jl < JCHUNK; ++jl) {
      const float4* row = (const float4*)(&tile[jl * 16]);  // (r,amp,r,amp) pairs
#pragma unroll
      for (int q = 0; q < 8; ++q) {
        const float4 ra = row[q];
        const float  p0 = ra.x * krev;
        accre[2 * q]     = fmaf(ra.y, __builtin_amdgcn_cosf(p0), accre[2 * q]);
        accim[2 * q]     = fmaf(ra.y, __builtin_amdgcn_sinf(p0), accim[2 * q]);
        const float  p1 = ra.z * krev;
        accre[2 * q + 1] = fmaf(ra.w, __builtin_amdgcn_cosf(p1), accre[2 * q + 1]);
        accim[2 * q + 1] = fmaf(ra.w, __builtin_amdgcn_sinf(p1), accim[2 * q + 1]);
      }
    }
    __syncthreads();
  }

  // out[b, i, 2m] = re, out[b, i, 2m+1] = im; zero padded targets.
  float2* ob = (float2*)out + ((size_t)b * N + i0) * NW;
#pragma unroll
  for (int il = 0; il < 16; ++il) {
    const float vm = (pb[i0 + il] != 0) ? 0.0f : 1.0f;
    ob[(size_t)il * NW + m] = make_float2(accre[il] * vm, accim[il] * vm);
  }
}

extern "C" void kernel_launch(void* const* d_in, const int* in_sizes, int n_in,
                              void* d_out, int out_size, void* d_ws, size_t ws_size,
                              hipStream_t stream) {
  const float*         coords = (const float*)d_in[0];
  const unsigned char* pad    = (const unsigned char*)d_in[1];  // jnp bool = 1 byte
  float*               out    = (float*)d_out;
  const int N = 2048;
  const int B = in_sizes[1] / N;   // key_padding_mask has B*N elements
  dim3 grid(N / 16, B);
  spatial_embed_kernel<<<grid, 256, 0, stream>>>(coords, pad, out, N);
}